// Warp_Object_76020921139524
// MI455X (gfx1250) — compile-verified
//
#include <hip/hip_runtime.h>
#include <stdint.h>

// Bicubic warp, MI455X (gfx1250). Memory-bound gather kernel:
//  - TDM (tensor_load_to_lds) stages a 36x33x3 haloed image tile into LDS
//    (one 3-D descriptor: 36 cols x 33 rows x 3 planes, plane stride 4*H*W)
//  - DMA issue happens FIRST; dx/dy loads + cubic coefficient math overlap the
//    in-flight DMA; s_wait_tensorcnt + barrier only afterwards
//  - 16-tap bicubic stencil gathered from LDS, coeffs amortized over the 3
//    channel planes sharing one displacement plane (reference tile-order quirk)
//  - border tiles / outlier displacements fall back to clamped global gathers

#define Bz   4
#define Cc   3
#define Hh   512
#define Ww   512
#define HW   (Hh * Ww)
#define TILE 16
#define HALO 8
#define RW   36              // staged columns  (16 + 2*8 halo + 4 slack)
#define RH   33              // staged rows     (16 + 2*8 halo + 1)
#define LDS_PLANE (RH * RW)

typedef uint32_t u32x4 __attribute__((ext_vector_type(4)));
typedef uint32_t u32x8 __attribute__((ext_vector_type(8)));

__device__ __forceinline__ void cubic_coeffs(float t, float c[4]) {
    float t2 = t * t;
    float t3 = t2 * t;
    c[0] = (-t3 + 2.0f * t2 - t) * 0.5f;
    c[1] = (3.0f * t3 - 5.0f * t2 + 2.0f) * 0.5f;
    c[2] = (-3.0f * t3 + 4.0f * t2 + t) * 0.5f;
    c[3] = 1.0f - (c[0] + c[1] + c[2]);
}

__global__ __launch_bounds__(256) void warp_bicubic_tdm(
    const float* __restrict__ img,   // [B*C, H, W]
    const float* __restrict__ dX,    // [B, H, W]
    const float* __restrict__ dY,    // [B, H, W]
    float* __restrict__ out)         // [B*C, H, W]
{
    __shared__ __align__(16) float sTile[Cc * LDS_PLANE];   // 14,256 B

    const int m  = blockIdx.z;                 // displacement plane (k % B)
    const int tx = blockIdx.x, ty = blockIdx.y;
    const int lx = threadIdx.x & 15;
    const int ly = threadIdx.x >> 4;
    const int X0 = tx * TILE, Y0 = ty * TILE;
    const int x  = X0 + lx,   y  = Y0 + ly;

    const int  gx0 = X0 - HALO, gy0 = Y0 - HALO;          // staged-region origin
    const bool interior = (tx >= 1) & (tx <= 30) & (ty >= 1) & (ty <= 30);

    // ---------------- stage 1: issue the TDM DMA (no wait yet) ----------------
    if (interior && threadIdx.x < 32) {   // wave 0 only (TDM ignores EXEC)
        uint64_t ga = (uint64_t)(uintptr_t)img +
            4ull * ((uint64_t)m * HW + (uint64_t)gy0 * Ww + (uint64_t)gx0);
        uint32_t lds_off = (uint32_t)(uintptr_t)(&sTile[0]);  // LDS byte offset

        u32x4 g0; u32x8 g1; u32x4 g2; u32x4 g3;
        // ---- D# group 0: count=1 valid descriptor, lds_addr, global_addr, type=2
        g0[0] = 1u;
        g0[1] = lds_off;
        g0[2] = (uint32_t)ga;
        g0[3] = (uint32_t)(ga >> 32) | 0x80000000u;   // type=2 in bits 127:126
        // ---- D# group 1
        g1[0] = (2u << 16);                 // workgroup_mask=0, data_size=4B, no pad/iter
        g1[1] = 0u;                         // tensor_dim0(=2^20) low16 -> 0
        g1[2] = 0x0010u;                    // tensor_dim0 hi16=0x10 | tensor_dim1 lo16=0
        g1[3] = 0x0010u | ((uint32_t)RW << 16);  // tensor_dim1 hi16 | tile_dim0=36
        g1[4] = (uint32_t)RH | (3u << 16);  // tile_dim1=33 | tile_dim2=3
        g1[5] = (uint32_t)Ww;               // tensor_dim0_stride = 512 (row stride)
        g1[6] = 0u;                         // stride0 hi16 | stride1 lo16 (both 0)
        g1[7] = (uint32_t)((4ull * HW) >> 16);  // tensor_dim1_stride = 4*H*W elems
        // ---- D# group 2: tensor_dim2=3, tensor_dim3=1, tile_dim3=0
        g2[0] = 3u; g2[1] = 1u; g2[2] = 0u; g2[3] = 0u;
        // ---- D# group 3: unused
        g3[0] = 0u; g3[1] = 0u; g3[2] = 0u; g3[3] = 0u;

        asm volatile("tensor_load_to_lds %0, %1, %2, %3"
                     :
                     : "s"(g0), "s"(g1), "s"(g2), "s"(g3)
                     : "memory");
    }

    // ---- stage 2: independent work overlaps the DMA --------------------------
    // per-pixel sampling coordinate (faithful to reference arithmetic)
    const float Wm1 = 511.0f, Hm1 = 511.0f;
    const size_t didx = (size_t)m * HW + (size_t)y * Ww + (size_t)x;
    const float dxv = dX[didx];
    const float dyv = dY[didx];

    float nX = 2.0f * ((float)x + dxv) / Wm1 - 1.0f;
    float nY = 2.0f * ((float)y + dyv) / Hm1 - 1.0f;
    float xm = (nX + 1.0f) * Wm1 * 0.5f;
    float ym = (nY + 1.0f) * Hm1 * 0.5f;

    float xf = floorf(xm), yf = floorf(ym);
    float cx[4], cy[4];
    cubic_coeffs(xm - xf, cx);
    cubic_coeffs(ym - yf, cy);
    int x0i = (int)xf, y0i = (int)yf;

    const int lx0 = x0i - gx0;          // region-local stencil origin
    const int ly0 = y0i - gy0;
    const bool fast = interior &
                      (lx0 >= 1) & (lx0 <= RW - 3) &
                      (ly0 >= 1) & (ly0 <= RH - 3);

    // ---- stage 3: complete the DMA, then gather ------------------------------
    if (interior) {
        if (threadIdx.x < 32) {
            __builtin_amdgcn_s_wait_tensorcnt(0);
        }
        __syncthreads();
    }

    if (fast) {
        // All 16 taps in the staged (clamp-free, interior) window: LDS gathers.
        #pragma unroll
        for (int ch = 0; ch < Cc; ++ch) {
            const float* P = &sTile[ch * LDS_PLANE + (ly0 - 1) * RW + (lx0 - 1)];
            float acc = 0.0f;
            #pragma unroll
            for (int j = 0; j < 4; ++j) {
                const float* R = P + j * RW;
                float rs = cx[0] * R[0] + cx[1] * R[1] + cx[2] * R[2] + cx[3] * R[3];
                acc += cy[j] * rs;
            }
            out[(size_t)(m + 4 * ch) * HW + (size_t)y * Ww + x] = acc;
        }
    } else {
        // Border tiles / rare large displacements: clamped global gather (L2-hot image).
        int xi[4], yi[4];
        #pragma unroll
        for (int i = 0; i < 4; ++i) {
            xi[i] = min(max(x0i + i - 1, 0), Ww - 1);
            yi[i] = min(max(y0i + i - 1, 0), Hh - 1);
        }
        #pragma unroll
        for (int ch = 0; ch < Cc; ++ch) {
            const float* P = img + (size_t)(m + 4 * ch) * HW;
            float acc = 0.0f;
            #pragma unroll
            for (int j = 0; j < 4; ++j) {
                const float* R = P + (size_t)yi[j] * Ww;
                float rs = cx[0] * R[xi[0]] + cx[1] * R[xi[1]] +
                           cx[2] * R[xi[2]] + cx[3] * R[xi[3]];
                acc += cy[j] * rs;
            }
            out[(size_t)(m + 4 * ch) * HW + (size_t)y * Ww + x] = acc;
        }
    }
}

extern "C" void kernel_launch(void* const* d_in, const int* in_sizes, int n_in,
                              void* d_out, int out_size, void* d_ws, size_t ws_size,
                              hipStream_t stream) {
    (void)in_sizes; (void)n_in; (void)out_size; (void)d_ws; (void)ws_size;
    const float* img = (const float*)d_in[0];   // [4,3,512,512]
    const float* dX  = (const float*)d_in[1];   // [4,1,512,512]
    const float* dY  = (const float*)d_in[2];   // [4,1,512,512]
    float* out       = (float*)d_out;           // [4,3,512,512]

    dim3 grid(Ww / TILE, Hh / TILE, Bz);        // 32 x 32 x 4
    dim3 block(256);                            // 16x16 tile, 8 wave32s
    warp_bicubic_tdm<<<grid, block, 0, stream>>>(img, dX, dY, out);
}